// BinarySphericalQuantizer_15917148799550
// MI455X (gfx1250) — compile-verified
//
#include <hip/hip_runtime.h>

// ---------------- problem geometry ----------------
#define EMBED   16
#define ROWS    (16 * 131072)            // 2,097,152 rows of 16 floats
#define BLOCKS  512
#define TPB     256
#define WAVES   (TPB / 32)               // 8 waves per block (wave32)
#define GSTRIDE (BLOCKS * TPB)           // 131072 rows per sweep
#define ITERS   (ROWS / GSTRIDE)         // 16
#define NPART   BLOCKS                   // partials per metric in d_ws
#define EPSV    1e-8f

// d_out layout (floats, concatenated in reference return order)
#define ZQ_N     ((size_t)ROWS * EMBED)          // 33,554,432
#define LOSS_OFF (ZQ_N)                          // 1 scalar
#define IDX_OFF  (LOSS_OFF + 1)                  // ROWS indices (as float)
#define AP_OFF   (IDX_OFF + (size_t)ROWS)        // 16 x 2 avg_prob

typedef float v8f __attribute__((ext_vector_type(8)));
typedef float v2f __attribute__((ext_vector_type(2)));

__global__ __launch_bounds__(TPB) void bsq_main(const float* __restrict__ z,
                                                float* __restrict__ out,
                                                float* __restrict__ ws) {
    __shared__ float S[WAVES * 32 * EMBED];   // per-wave 32x16 transpose staging
    __shared__ float wtot[WAVES * EMBED];     // per-wave per-dim sums
    __shared__ float whent[WAVES];            // per-wave entropy sums

    const int tid  = blockIdx.x * TPB + threadIdx.x;
    const int lane = threadIdx.x & 31;
    const int wv   = threadIdx.x >> 5;

    float sacc[EMBED];
#pragma unroll
    for (int i = 0; i < EMBED; ++i) sacc[i] = 0.0f;
    float hacc = 0.0f;

    for (int it = 0; it < ITERS; ++it) {
        const int r = tid + it * GSTRIDE;     // uniform across wave, always < ROWS
        const float4* zr = (const float4*)(z + (size_t)r * EMBED);
        float4 a0 = zr[0], a1 = zr[1], a2 = zr[2], a3 = zr[3];
        if (it + 1 < ITERS)
            __builtin_prefetch(z + (size_t)(r + GSTRIDE) * EMBED, 0, 0); // global_prefetch_b8

        float zv[EMBED];
        zv[0]=a0.x; zv[1]=a0.y; zv[2]=a0.z; zv[3]=a0.w;
        zv[4]=a1.x; zv[5]=a1.y; zv[6]=a1.z; zv[7]=a1.w;
        zv[8]=a2.x; zv[9]=a2.y; zv[10]=a2.z; zv[11]=a2.w;
        zv[12]=a3.x; zv[13]=a3.y; zv[14]=a3.z; zv[15]=a3.w;

        float n2 = 0.0f;
#pragma unroll
        for (int i = 0; i < EMBED; ++i) n2 = fmaf(zv[i], zv[i], n2);
        // 1/max(||z||,1e-12) == rsqrt(max(||z||^2, 1e-24))
        const float scale = rsqrtf(fmaxf(n2, 1e-24f));

        int idx = 0;
        float q[EMBED];
#pragma unroll
        for (int i = 0; i < EMBED; ++i) {
            const float zn = zv[i] * scale;
            const float p  = 1.0f / (1.0f + __expf(zn));    // sigmoid(-zn)
            const float om = 1.0f - p;
            hacc -= p * __logf(p + EPSV) + om * __logf(om + EPSV);
            sacc[i] += p;
            const bool b = zn > 0.0f;
            idx |= (b ? 1 : 0) << (15 - i);
            q[i] = b ? 0.25f : -0.25f;                      // sign(z)/sqrt(16)
        }

        float4* qo = (float4*)(out + (size_t)r * EMBED);
        qo[0] = make_float4(q[0], q[1], q[2], q[3]);
        qo[1] = make_float4(q[4], q[5], q[6], q[7]);
        qo[2] = make_float4(q[8], q[9], q[10], q[11]);
        qo[3] = make_float4(q[12], q[13], q[14], q[15]);
        out[IDX_OFF + (size_t)r] = (float)idx;
    }

    // ---- per-wave cross-lane reduction of sacc[16] via exact f32 WMMA ----
    // Stage the wave's 32x16 partial matrix in LDS, then sum its columns with
    // V_WMMA_F32_16X16X4_F32 using A = ones(16x4): D[m][n] = sum_k B[k][n] + C[m][n].
    float* Sw = S + wv * (32 * EMBED);
#pragma unroll
    for (int i = 0; i < EMBED; ++i) Sw[lane * EMBED + i] = sacc[i];
    __syncthreads();

    v2f aones = {1.0f, 1.0f};
    v8f c = {0.0f, 0.0f, 0.0f, 0.0f, 0.0f, 0.0f, 0.0f, 0.0f};
    const int col = lane & 15;
    const int khi = (lane >> 4) << 1;          // B layout: vgpr0={K0,K2}, vgpr1={K1,K3}
#pragma unroll
    for (int ch = 0; ch < 8; ++ch) {           // 8 chunks of K=4 cover 32 source lanes
        const int k0 = ch * 4 + khi;
        v2f b;
        b[0] = Sw[(k0 + 0) * EMBED + col];
        b[1] = Sw[(k0 + 1) * EMBED + col];
        c = __builtin_amdgcn_wmma_f32_16x16x4_f32(false, aones, false, b,
                                                  (short)0, c, false, false);
    }

    // scalar entropy: wave32 shuffle-xor tree
    float h = hacc;
#pragma unroll
    for (int off = 16; off > 0; off >>= 1) h += __shfl_xor(h, off, 32);

    // C row M=0 lives in c[0] on lanes 0..15 (N = lane)
    if (lane < EMBED) wtot[wv * EMBED + lane] = c[0];
    if (lane == 0)    whent[wv] = h;
    __syncthreads();

    // block-level combine (wave 0), write deterministic partials to d_ws
    if (threadIdx.x < 32) {
        if (lane < EMBED) {
            float s = 0.0f;
#pragma unroll
            for (int w = 0; w < WAVES; ++w) s += wtot[w * EMBED + lane];
            ws[lane * NPART + blockIdx.x] = s;
        } else if (lane == 16) {
            float s = 0.0f;
#pragma unroll
            for (int w = 0; w < WAVES; ++w) s += whent[w];
            ws[16 * NPART + blockIdx.x] = s;
        }
    }
}

__global__ __launch_bounds__(256) void bsq_finalize(const float* __restrict__ ws,
                                                    float* __restrict__ out) {
    __shared__ float tot[17];
    __shared__ float pw[WAVES];
    const int t    = threadIdx.x;
    const int lane = t & 31;
    const int wv   = t >> 5;

    for (int m = 0; m < 17; ++m) {
        float a = ws[m * NPART + t] + ws[m * NPART + t + 256];  // NPART == 512
#pragma unroll
        for (int off = 16; off > 0; off >>= 1) a += __shfl_xor(a, off, 32);
        if (lane == 0) pw[wv] = a;
        __syncthreads();
        if (t == 0) {
            float s = 0.0f;
            for (int w = 0; w < WAVES; ++w) s += pw[w];
            tot[m] = s;
        }
        __syncthreads();
    }

    const float invN = 1.0f / (float)ROWS;
    if (t < EMBED) {
        const float ap = tot[t] * invN;
        out[AP_OFF + 2 * t]     = ap;
        out[AP_OFF + 2 * t + 1] = 1.0f - ap;
    }
    if (t == 0) {
        float cb = 0.0f;
        for (int i = 0; i < EMBED; ++i) {
            const float ap = tot[i] * invN;
            const float om = 1.0f - ap;
            cb -= ap * __logf(ap + EPSV) + om * __logf(om + EPSV);
        }
        const float persample = tot[16] * invN;
        out[LOSS_OFF] = persample - cb;   // gamma0 = gamma = zeta = inv_t = 1
    }
}

extern "C" void kernel_launch(void* const* d_in, const int* in_sizes, int n_in,
                              void* d_out, int out_size, void* d_ws, size_t ws_size,
                              hipStream_t stream) {
    const float* z = (const float*)d_in[0];
    float* out = (float*)d_out;
    float* ws  = (float*)d_ws;    // 17 * 512 floats of deterministic partials

    bsq_main<<<BLOCKS, TPB, 0, stream>>>(z, out, ws);
    bsq_finalize<<<1, 256, 0, stream>>>(ws, out);
}